// KroneSoftmax_59485297049801
// MI455X (gfx1250) — compile-verified
//
#include <hip/hip_runtime.h>

typedef __attribute__((ext_vector_type(2))) float v2f;
typedef __attribute__((ext_vector_type(8))) float v8f;

// One workgroup (256 threads = 8 wave32) per batch row.
// Row layout: a = x[0][row][0:256], b = x[1][row][0:256].
// y[i*256+j] = a[i]*b[j]; out = softmax(y) over 65536 elements.
__global__ __launch_bounds__(256)
void KroneSoftmax_kernel(const float* __restrict__ x,
                         float* __restrict__ out,
                         int batch) {
  const int row  = blockIdx.x;
  const int tid  = threadIdx.x;
  const int lane = tid & 31;
  const int wave = tid >> 5;

  __shared__ float sA[256];
  __shared__ float sB[256];
  __shared__ float sRed[4][8];   // per-wave maxA/minA/maxB/minB
  __shared__ float sSum[8];      // per-wave exp-sum partials

  const float* a_ptr = x + (size_t)row * 256;
  const float* b_ptr = x + (size_t)batch * 256 + (size_t)row * 256;

  float av = a_ptr[tid];
  float bv = b_ptr[tid];
  sA[tid] = av;
  sB[tid] = bv;

  // ---- Exact max of all products via factorized min/max reduction ----
  float maxA = av, minA = av, maxB = bv, minB = bv;
#pragma unroll
  for (int off = 16; off > 0; off >>= 1) {
    maxA = fmaxf(maxA, __shfl_xor(maxA, off, 32));
    minA = fminf(minA, __shfl_xor(minA, off, 32));
    maxB = fmaxf(maxB, __shfl_xor(maxB, off, 32));
    minB = fminf(minB, __shfl_xor(minB, off, 32));
  }
  if (lane == 0) {
    sRed[0][wave] = maxA; sRed[1][wave] = minA;
    sRed[2][wave] = maxB; sRed[3][wave] = minB;
  }
  __syncthreads();
  maxA = sRed[0][0]; minA = sRed[1][0]; maxB = sRed[2][0]; minB = sRed[3][0];
#pragma unroll
  for (int w = 1; w < 8; ++w) {
    maxA = fmaxf(maxA, sRed[0][w]); minA = fminf(minA, sRed[1][w]);
    maxB = fmaxf(maxB, sRed[2][w]); minB = fminf(minB, sRed[3][w]);
  }
  const float M = fmaxf(fmaxf(maxA * maxB, maxA * minB),
                        fmaxf(minA * maxB, minA * minB));

  // ---- Pass 1: S = sum exp(a_i*b_j - M), tiles via WMMA rank-1 GEMM ----
  // Wave w owns a-chunks {2w, 2w+1} x all 16 b-chunks = 32 tiles of 16x16.
  float lsum = 0.0f;
  for (int ca = wave * 2; ca < wave * 2 + 2; ++ca) {
    v2f Areg;
    Areg.x = (lane < 16) ? sA[ca * 16 + lane] : 0.0f;  // A: M=lane, K=0
    Areg.y = 0.0f;                                     // K=1..3 zero
    for (int cb = 0; cb < 16; ++cb) {
      v2f Breg;
      Breg.x = (lane < 16) ? sB[cb * 16 + lane] : 0.0f; // B: K=0, N=lane
      Breg.y = 0.0f;
      v8f c = {};
      v8f d = __builtin_amdgcn_wmma_f32_16x16x4_f32(
          false, Areg, false, Breg, (short)0, c, false, false);
#pragma unroll
      for (int v = 0; v < 8; ++v) lsum += __expf(d[v] - M);
    }
  }
#pragma unroll
  for (int off = 16; off > 0; off >>= 1) lsum += __shfl_xor(lsum, off, 32);
  if (lane == 0) sSum[wave] = lsum;
  __syncthreads();
  float S = 0.0f;
#pragma unroll
  for (int w = 0; w < 8; ++w) S += sSum[w];
  const float invS = 1.0f / S;

  // ---- Pass 2: recompute tiles, write exp(d-M)*invS exactly once ----
  float* orow = out + (size_t)row * 65536;
  for (int ca = wave * 2; ca < wave * 2 + 2; ++ca) {
    v2f Areg;
    Areg.x = (lane < 16) ? sA[ca * 16 + lane] : 0.0f;
    Areg.y = 0.0f;
    for (int cb = 0; cb < 16; ++cb) {
      v2f Breg;
      Breg.x = (lane < 16) ? sB[cb * 16 + lane] : 0.0f;
      Breg.y = 0.0f;
      v8f c = {};
      v8f d = __builtin_amdgcn_wmma_f32_16x16x4_f32(
          false, Areg, false, Breg, (short)0, c, false, false);
      // D layout: element (M = v + 8*(lane>=16), N = lane%16)
      const int i0 = ca * 16 + ((lane >= 16) ? 8 : 0);
      const int j  = cb * 16 + (lane & 15);
#pragma unroll
      for (int v = 0; v < 8; ++v) {
        orow[(size_t)(i0 + v) * 256 + j] = __expf(d[v] - M) * invS;
      }
    }
  }
}

extern "C" void kernel_launch(void* const* d_in, const int* in_sizes, int n_in,
                              void* d_out, int out_size, void* d_ws, size_t ws_size,
                              hipStream_t stream) {
  (void)n_in; (void)d_ws; (void)ws_size; (void)out_size;
  const float* x = (const float*)d_in[0];
  float* out = (float*)d_out;
  // x is [2, batch, 256] flattened
  const int batch = in_sizes[0] / (2 * 256);
  KroneSoftmax_kernel<<<batch, 256, 0, stream>>>(x, out, batch);
}